// MarginRankingLoss_90460601188750
// MI455X (gfx1250) — compile-verified
//
#include <hip/hip_runtime.h>
#include <math.h>

typedef __attribute__((ext_vector_type(2))) float v2f;
typedef __attribute__((ext_vector_type(8))) float v8f;

#define MARGIN 0.1f
#define EPSV   1e-6f

// One 16x16 tile of the pair matrix per wave iteration.
// dp/dt tiles via V_WMMA_F32_16X16X4_F32 as exact rank-2 outer products:
//   dp = A x B,  A row m = [p_{i0+m}, 1, 0, 0],  B col n = [1, -p_{j0+n}, 0, 0]^T
template <bool ALIGNED>
__global__ void margin_rank_pairs_kernel(const float* __restrict__ p,
                                         const float* __restrict__ t,
                                         const unsigned char* __restrict__ mask,
                                         float* __restrict__ blk_sum,
                                         unsigned int* __restrict__ blk_cnt,
                                         int n, long long total_tiles,
                                         int waves_total) {
    __shared__ float    s_sum[8];
    __shared__ unsigned s_cnt[8];

    const int tid   = threadIdx.x;
    const int lane  = tid & 31;
    const int wave  = tid >> 5;
    const int nwav  = blockDim.x >> 5;
    const int wglob = blockIdx.x * nwav + wave;

    const int  half    = lane >> 4;     // 0: lanes 0-15, 1: lanes 16-31
    const int  l16     = lane & 15;
    const bool lowhalf = (half == 0);

    float    lsum = 0.0f;
    unsigned lcnt = 0u;

    for (long long k = wglob; k < total_tiles; k += waves_total) {
        // Decode triangular linear index k -> (ti <= tj). Wave-uniform.
        int tj = (int)((sqrt(8.0 * (double)k + 1.0) - 1.0) * 0.5);
        while ((long long)(tj + 1) * (tj + 2) / 2 <= k) ++tj;
        while ((long long)tj * (tj + 1) / 2 > k) --tj;
        const int ti = (int)(k - (long long)tj * (tj + 1) / 2);
        const int i0 = ti * 16;
        const int j0 = tj * 16;

        // ---- Operand gathers (unconditional in the ALIGNED fast path) ----
        float pa, ta, pb, tb;
        unsigned rowlo, rowhi;          // 8 row-mask bytes
        bool     mj;
        const int ibase = i0 + 8 * half;
        const int j     = j0 + l16;

        if (ALIGNED) {
            const int ia = i0 + l16;
            pa = p[ia]; ta = t[ia];
            pb = p[j];  tb = t[j];
            const uint2 m8 = *(const uint2*)(mask + ibase);   // 8-byte aligned
            rowlo = m8.x; rowhi = m8.y;
            mj = (mask[j] != 0);
        } else {
            const int nm1 = n - 1;
            const int ia = min(i0 + l16, nm1);
            const int jc = min(j, nm1);
            pa = p[ia]; ta = t[ia];
            pb = p[jc]; tb = t[jc];
            unsigned lo = 0, hi = 0;
            #pragma unroll
            for (int r = 0; r < 4; ++r) {
                lo |= ((unsigned)mask[min(ibase + r,     nm1)]) << (8 * r);
                hi |= ((unsigned)mask[min(ibase + 4 + r, nm1)]) << (8 * r);
            }
            rowlo = lo; rowhi = hi;
            mj = (j < n) && (mask[jc] != 0);
        }

        // A (16x4 f32, 2 VGPRs): lanes 0-15 -> K=0,1 ; lanes 16-31 -> K=2,3 (zero)
        v2f aP; aP.x = lowhalf ? pa   : 0.0f; aP.y = lowhalf ? 1.0f : 0.0f;
        v2f aT; aT.x = lowhalf ? ta   : 0.0f; aT.y = lowhalf ? 1.0f : 0.0f;
        // B (4x16 f32, 2 VGPRs): lanes 0-15 -> K=0,1 ; lanes 16-31 -> K=2,3 (zero)
        v2f bP; bP.x = lowhalf ? 1.0f : 0.0f; bP.y = lowhalf ? -pb : 0.0f;
        v2f bT; bT.x = lowhalf ? 1.0f : 0.0f; bT.y = lowhalf ? -tb : 0.0f;

        v8f zero = {};
        v8f dp = __builtin_amdgcn_wmma_f32_16x16x4_f32(false, aP, false, bP,
                                                       (short)0, zero, false, false);
        v8f dt = __builtin_amdgcn_wmma_f32_16x16x4_f32(false, aT, false, bT,
                                                       (short)0, zero, false, false);

        // D layout: VGPR r, lane -> element (m = r + 8*(lane/16), col = lane%16)
        #pragma unroll
        for (int r = 0; r < 8; ++r) {
            const int   i    = ibase + r;
            const float dpv  = dp[r];
            const float dtv  = dt[r];
            const unsigned mb = ((r < 4 ? rowlo : rowhi) >> (8 * (r & 3))) & 0xFFu;
            bool valid = (i < j) && mj && (mb != 0u) && (fabsf(dtv) >= EPSV);
            if (!ALIGNED) valid = valid && (i < n);
            // -sign(dt)*dp ; dt==0 excluded by |dt|>=eps
            const float s = (dtv > 0.0f) ? -dpv : dpv;
            const float h = fmaxf(0.0f, s + MARGIN);
            lsum += valid ? h : 0.0f;
            lcnt += valid ? 1u : 0u;
        }
    }

    // Wave32 reduction
    #pragma unroll
    for (int off = 16; off > 0; off >>= 1) {
        lsum += __shfl_xor(lsum, off, 32);
        lcnt += __shfl_xor(lcnt, off, 32);
    }
    if (lane == 0) { s_sum[wave] = lsum; s_cnt[wave] = lcnt; }
    __syncthreads();

    if (tid == 0) {
        float    bsum = 0.0f;
        unsigned bcnt = 0u;
        for (int w = 0; w < nwav; ++w) { bsum += s_sum[w]; bcnt += s_cnt[w]; }
        blk_sum[blockIdx.x] = bsum;   // written unconditionally every launch
        blk_cnt[blockIdx.x] = bcnt;
    }
}

__global__ void margin_rank_finalize_kernel(const float* __restrict__ blk_sum,
                                            const unsigned int* __restrict__ blk_cnt,
                                            float* __restrict__ out, int nblk) {
    __shared__ float    ss[256];
    __shared__ unsigned sc[256];
    float    a = 0.0f;
    unsigned c = 0u;
    for (int i = threadIdx.x; i < nblk; i += 256) { a += blk_sum[i]; c += blk_cnt[i]; }
    ss[threadIdx.x] = a; sc[threadIdx.x] = c;
    __syncthreads();
    for (int off = 128; off > 0; off >>= 1) {
        if ((int)threadIdx.x < off) {
            ss[threadIdx.x] += ss[threadIdx.x + off];
            sc[threadIdx.x] += sc[threadIdx.x + off];
        }
        __syncthreads();
    }
    if (threadIdx.x == 0) {
        const unsigned cnt = sc[0];
        out[0] = ss[0] / (float)(cnt == 0u ? 1u : cnt);
    }
}

extern "C" void kernel_launch(void* const* d_in, const int* in_sizes, int n_in,
                              void* d_out, int out_size, void* d_ws, size_t ws_size,
                              hipStream_t stream) {
    const float*         p    = (const float*)d_in[0];
    const float*         t    = (const float*)d_in[1];
    const unsigned char* mask = (const unsigned char*)d_in[2];
    float*               out  = (float*)d_out;

    const int n = in_sizes[0];
    const int NT = (n + 15) / 16;
    const long long total_tiles = (long long)NT * (NT + 1) / 2;

    int nblk = 1024;
    const size_t per_blk = sizeof(float) + sizeof(unsigned);
    if (ws_size < (size_t)nblk * per_blk) {
        nblk = (int)(ws_size / per_blk);
        if (nblk < 1) nblk = 1;
    }

    float*        bs = (float*)d_ws;
    unsigned int* bc = (unsigned int*)(bs + nblk);

    const int threads = 256;
    const int waves_total = nblk * (threads / 32);

    if ((n & 15) == 0) {
        margin_rank_pairs_kernel<true><<<nblk, threads, 0, stream>>>(
            p, t, mask, bs, bc, n, total_tiles, waves_total);
    } else {
        margin_rank_pairs_kernel<false><<<nblk, threads, 0, stream>>>(
            p, t, mask, bs, bc, n, total_tiles, waves_total);
    }
    margin_rank_finalize_kernel<<<1, 256, 0, stream>>>(bs, bc, out, nblk);
}